// GraphConvolution_38774964748853
// MI455X (gfx1250) — compile-verified
//
#include <hip/hip_runtime.h>

#define N_NODES 100000
#define N_EDGES 1600000
#define HIDDEN  128
#define LN_EPS  1e-5f
#define LDSP    132   // padded LDS row stride (floats); 132 % 64 == 4 -> conflict-free

typedef float v2f __attribute__((ext_vector_type(2)));
typedef float v8f __attribute__((ext_vector_type(8)));

// ---------------- kernel 1: zero the aggregation buffer ----------------
__global__ void gc_zero_kernel(float* __restrict__ p, int n4) {
    int i = blockIdx.x * blockDim.x + threadIdx.x;
    if (i < n4) ((float4*)p)[i] = make_float4(0.f, 0.f, 0.f, 0.f);
}

// ---------------- kernel 2: edge gather + scatter-add (segment sum) ----
// One wave32 per edge: each lane moves 4 consecutive floats (float4 load,
// 4x f32 atomic adds). h and agg both fit in the 192MB L2, so the atomics
// resolve at the L2 atomic units.
__global__ void gc_edge_scatter_kernel(const float* __restrict__ h,
                                       const int* __restrict__ src,
                                       const int* __restrict__ dst,
                                       float* __restrict__ agg) {
    int gt   = blockIdx.x * blockDim.x + threadIdx.x;
    int e    = gt >> 5;           // wave-uniform edge id
    int lane = threadIdx.x & 31;
    if (e >= N_EDGES) return;
    int s = src[e];
    int d = dst[e];
    float4 v = ((const float4*)(h + (size_t)s * HIDDEN))[lane];
    float* ap = agg + (size_t)d * HIDDEN + lane * 4;
    atomicAdd(ap + 0, v.x);
    atomicAdd(ap + 1, v.y);
    atomicAdd(ap + 2, v.z);
    atomicAdd(ap + 3, v.w);
}

// ---------------- kernel 3: fused WMMA GEMM + ReLU + residual + LN -----
// Block = 128 threads (4 waves), handles 16 nodes (one M-tile).
// x = relu(agg @ W^T) + h ; LayerNorm over HIDDEN.
// A = agg tile (16 x 128, M x K), B[k][j] = W[j][k] (4 x 16 per step).
__global__ void __launch_bounds__(128)
gc_gemm_ln_kernel(const float* __restrict__ agg,
                  const float* __restrict__ W,
                  const float* __restrict__ h,
                  const float* __restrict__ gamma,
                  const float* __restrict__ beta,
                  float* __restrict__ out) {
    __shared__ float As[16 * LDSP];
    __shared__ float Xs[16 * LDSP];

    const int tid  = threadIdx.x;
    const int row0 = blockIdx.x * 16;

    // Stage A tile (16 x 128 f32) into LDS, coalesced float4 loads.
    for (int i = tid; i < 16 * 32; i += 128) {
        int r  = i >> 5;
        int c4 = i & 31;
        float4 v = ((const float4*)(agg + (size_t)(row0 + r) * HIDDEN))[c4];
        float* dp = &As[r * LDSP + c4 * 4];
        dp[0] = v.x; dp[1] = v.y; dp[2] = v.z; dp[3] = v.w;
    }
    __syncthreads();

    const int wave = tid >> 5;
    const int lane = tid & 31;
    const int lr   = lane & 15;   // M (for A) / N (for B,C/D) index
    const int hi   = lane >> 4;   // K-half selector for A/B fragments

    // Each wave computes two 16-column output tiles.
    #pragma unroll
    for (int t = 0; t < 2; ++t) {
        const int jt = wave * 2 + t;
        const int j0 = jt * 16;
        v8f acc = {};

        #pragma unroll 4
        for (int kb = 0; kb < HIDDEN; kb += 4) {
            // A fragment: lane<16 -> a[M=lr][kb..kb+1]; lane>=16 -> a[lr][kb+2..kb+3]
            const float* ap = &As[lr * LDSP + kb + 2 * hi];
            v2f a; a.x = ap[0]; a.y = ap[1];
            // B fragment: B[k][j] = W[j][k]
            const float* bp = &W[(size_t)(j0 + lr) * HIDDEN + kb + 2 * hi];
            v2f b; b.x = bp[0]; b.y = bp[1];
            acc = __builtin_amdgcn_wmma_f32_16x16x4_f32(
                false, a, false, b, (short)0, acc, false, false);
        }

        // C/D layout: lane holds column N = lr, rows M = v + 8*hi.
        #pragma unroll
        for (int v = 0; v < 8; ++v) {
            int m = v + 8 * hi;
            float val = acc[v];
            Xs[m * LDSP + j0 + lr] = val > 0.f ? val : 0.f;  // ReLU
        }
    }
    __syncthreads();

    // LayerNorm: each wave handles 4 rows; 4 elements per lane per row.
    for (int rr = 0; rr < 4; ++rr) {
        const int r    = wave * 4 + rr;
        const int node = row0 + r;
        const float* hp = h + (size_t)node * HIDDEN;

        float x0 = Xs[r * LDSP + lane +  0] + hp[lane +  0];
        float x1 = Xs[r * LDSP + lane + 32] + hp[lane + 32];
        float x2 = Xs[r * LDSP + lane + 64] + hp[lane + 64];
        float x3 = Xs[r * LDSP + lane + 96] + hp[lane + 96];

        float s = x0 + x1 + x2 + x3;
        #pragma unroll
        for (int m = 16; m >= 1; m >>= 1) s += __shfl_xor(s, m, 32);
        float mu = s * (1.0f / HIDDEN);

        float d0 = x0 - mu, d1 = x1 - mu, d2 = x2 - mu, d3 = x3 - mu;
        float q = d0 * d0 + d1 * d1 + d2 * d2 + d3 * d3;
        #pragma unroll
        for (int m = 16; m >= 1; m >>= 1) q += __shfl_xor(q, m, 32);
        float inv = rsqrtf(q * (1.0f / HIDDEN) + LN_EPS);

        float* op = out + (size_t)node * HIDDEN;
        op[lane +  0] = d0 * inv * gamma[lane +  0] + beta[lane +  0];
        op[lane + 32] = d1 * inv * gamma[lane + 32] + beta[lane + 32];
        op[lane + 64] = d2 * inv * gamma[lane + 64] + beta[lane + 64];
        op[lane + 96] = d3 * inv * gamma[lane + 96] + beta[lane + 96];
    }
}

extern "C" void kernel_launch(void* const* d_in, const int* in_sizes, int n_in,
                              void* d_out, int out_size, void* d_ws, size_t ws_size,
                              hipStream_t stream) {
    const float* h     = (const float*)d_in[0];
    const float* W     = (const float*)d_in[1];
    const float* gamma = (const float*)d_in[2];
    const float* beta  = (const float*)d_in[3];
    const int*   src   = (const int*)d_in[4];
    const int*   dst   = (const int*)d_in[5];
    float*       out   = (float*)d_out;
    float*       agg   = (float*)d_ws;   // N_NODES * HIDDEN floats = 51.2 MB

    const int n4 = N_NODES * HIDDEN / 4;
    gc_zero_kernel<<<(n4 + 255) / 256, 256, 0, stream>>>(agg, n4);

    const long long nthreads = (long long)N_EDGES * 32;
    gc_edge_scatter_kernel<<<(int)((nthreads + 255) / 256), 256, 0, stream>>>(
        h, src, dst, agg);

    gc_gemm_ln_kernel<<<N_NODES / 16, 128, 0, stream>>>(
        agg, W, h, gamma, beta, out);
}